// AttentionLayer_wg_87694642249854
// MI455X (gfx1250) — compile-verified
//
#include <hip/hip_runtime.h>

typedef __attribute__((ext_vector_type(2))) float v2f;
typedef __attribute__((ext_vector_type(8))) float v8f;

#define HID 256
#define NTYPE 8
#define NHEADS 8

// ---------------- K0: zero the column-sum accumulator ----------------
__global__ void zero_kernel(float* __restrict__ colsum) {
    colsum[threadIdx.x] = 0.0f;
}

// ---------------- K1: column sums of feats (the 102 MB streaming pass) ----
// thread = column (256), block strides over rows -> fully coalesced 1KB rows
__global__ void colsum_kernel(const float* __restrict__ feats,
                              float* __restrict__ colsum, int nnodes) {
    const int col = threadIdx.x;
    float s = 0.0f;
    for (int r = blockIdx.x; r < nnodes; r += gridDim.x)
        s += feats[(size_t)r * HID + col];
    atomicAdd(&colsum[col], s);
}

// ---------------- K2: finalize mean + build Wt[128][256] -------------------
// Wt[c][d]: c in [0,64)  -> a[n][h][d]       (self weights,  c = n*8+h)
//           c in [64,128)-> a[n][h][256+d]   (other weights, c-64 = n*8+h)
__global__ void build_wt_mean(const float* __restrict__ a,
                              const float* __restrict__ colsum,
                              float* __restrict__ mean,
                              float* __restrict__ Wt, int nnodes) {
    const int c = blockIdx.x;
    const int t = threadIdx.x;   // 0..255
    if (c < 128) {
        const int cc = c & 63;                  // n*8 + h
        const int off = (c < 64) ? t : (HID + t);
        Wt[c * HID + t] = a[cc * (3 * HID) + off];
    } else {
        mean[t] = colsum[t] / (float)nnodes;
    }
}

// ---------------- K2b: R[n,h] = mean . a[n][h][512:768] --------------------
__global__ void build_R(const float* __restrict__ a,
                        const float* __restrict__ mean,
                        float* __restrict__ R) {
    __shared__ float red[256];
    const int nh = blockIdx.x;   // 0..63
    const int t = threadIdx.x;
    red[t] = mean[t] * a[nh * (3 * HID) + 2 * HID + t];
    __syncthreads();
    for (int s = 128; s > 0; s >>= 1) {
        if (t < s) red[t] += red[t + s];
        __syncthreads();
    }
    if (t == 0) R[nh] = red[0];
}

// ---------------- K3: gather + WMMA GEMM + softmax epilogue ----------------
// Block: 128 threads (4 waves), handles 8 subgraphs (64 X-rows).
// Wave w owns rows lr = 16*w + laneLo (A-matrix M dim), computes a 16x128 tile
// P[lr][col] = X[lr] . Wt[col] with 8 accumulators of V_WMMA_F32_16X16X4_F32.
__global__ __launch_bounds__(128) void attn_main(
    const float* __restrict__ feats, const int* __restrict__ subgraphs,
    const float* __restrict__ attentions, const float* __restrict__ Wt,
    const float* __restrict__ R, float* __restrict__ out, int Bv) {

    __shared__ float Plds[64 * 132];   // 64 rows x 128 cols, padded stride 132

    const int tid    = threadIdx.x;
    const int wid    = tid >> 5;
    const int lane   = tid & 31;
    const int laneLo = lane & 15;
    const int laneHi = lane >> 4;
    const int b_base = blockIdx.x * 8;

    // A-fragment row pointer for this lane: row lr = 16*wid + laneLo
    const int lr = 16 * wid + laneLo;
    int b = b_base + (lr >> 3);
    if (b >= Bv) b = Bv - 1;                 // safe clamp (writes are guarded)
    const int node = subgraphs[b * NTYPE + (lr & 7)];
    const float* rowA = feats + (size_t)node * HID;

    v8f acc[8] = {};

    for (int k0 = 0; k0 < HID; k0 += 4) {
        const int kk = k0 + 2 * laneHi;
        // A 16x4 f32 layout: lanes 0-15 hold K = k0..k0+1, lanes 16-31 K = k0+2..k0+3
        const v2f afr = *(const v2f*)(rowA + kk);
#pragma unroll
        for (int ct = 0; ct < 8; ++ct) {
            const int col = 16 * ct + laneLo;
            // B 4x16 f32 layout: VGPR v = K (v + 2*laneHi), N = laneLo -> contiguous in Wt[col]
            const v2f bfr = *(const v2f*)(Wt + col * HID + kk);
            acc[ct] = __builtin_amdgcn_wmma_f32_16x16x4_f32(
                false, afr, false, bfr, (short)0, acc[ct], false, false);
        }
    }

    // Spill C tiles: VGPR p -> row M = p (lanes 0-15) / p+8 (lanes 16-31), col = laneLo
#pragma unroll
    for (int ct = 0; ct < 8; ++ct) {
#pragma unroll
        for (int p = 0; p < 8; ++p) {
            Plds[(16 * wid + p + 8 * laneHi) * 132 + 16 * ct + laneLo] = acc[ct][p];
        }
    }
    __syncthreads();

    // Epilogue: 8 lb x 8 n x 8 h = 512 (b,n,h) tasks; softmax over the 7 partners
    for (int t = tid; t < 512; t += 128) {
        const int lb = t >> 6;
        const int n  = (t >> 3) & 7;
        const int h  = t & 7;
        const int bb = b_base + lb;
        if (bb >= Bv) continue;

        const float S    = Plds[(lb * 8 + n) * 132 + (n * 8 + h)];
        const float Rv   = R[n * 8 + h];
        const int   colO = 64 + n * 8 + h;

        float e[7];
        float mx = -3.4e38f;
#pragma unroll
        for (int k = 0; k < 7; ++k) {
            const int j = k + (k >= n ? 1 : 0);       // idx_others[n][k]
            float att = S + Plds[(lb * 8 + j) * 132 + colO] + Rv;
            att = att > 0.0f ? att : 0.2f * att;      // LeakyReLU(0.2)
            e[k] = att;
            mx = fmaxf(mx, att);
        }
        float sum = 0.0f;
#pragma unroll
        for (int k = 0; k < 7; ++k) {
            e[k] = __expf(e[k] - mx);
            sum += e[k];
        }
        const float inv = 1.0f / sum;
        const size_t obase = ((size_t)n * Bv + bb) * (NHEADS * 7) + (size_t)h * 7;
#pragma unroll
        for (int k = 0; k < 7; ++k) {
            out[obase + k] = 0.5f * (e[k] * inv) + 0.5f * attentions[obase + k];
        }
    }
}

// ---------------- launch ---------------------------------------------------
extern "C" void kernel_launch(void* const* d_in, const int* in_sizes, int n_in,
                              void* d_out, int out_size, void* d_ws, size_t ws_size,
                              hipStream_t stream) {
    const float* feats      = (const float*)d_in[0];
    const float* a          = (const float*)d_in[1];
    const float* attentions = (const float*)d_in[2];
    const int*   subgraphs  = (const int*)d_in[3];
    float*       out        = (float*)d_out;
    float*       ws         = (float*)d_ws;

    const int nnodes = in_sizes[0] / HID;
    const int Bv     = in_sizes[3] / NTYPE;

    float* colsum = ws;            // 256 floats
    float* mean   = ws + 256;      // 256 floats
    float* Rbuf   = ws + 512;      // 64 floats
    float* Wt     = ws + 1024;     // 128*256 floats  (~132 KB total ws use)

    hipLaunchKernelGGL(zero_kernel,   dim3(1),    dim3(256), 0, stream, colsum);
    hipLaunchKernelGGL(colsum_kernel, dim3(1024), dim3(256), 0, stream, feats, colsum, nnodes);
    hipLaunchKernelGGL(build_wt_mean, dim3(129),  dim3(256), 0, stream, a, colsum, mean, Wt, nnodes);
    hipLaunchKernelGGL(build_R,       dim3(64),   dim3(256), 0, stream, a, mean, Rbuf);

    const int nblocks = (Bv + 7) / 8;
    hipLaunchKernelGGL(attn_main, dim3(nblocks), dim3(128), 0, stream,
                       feats, subgraphs, attentions, Wt, Rbuf, out, Bv);
}